// GraphSAGE_52690658788080
// MI455X (gfx1250) — compile-verified
//
#include <hip/hip_runtime.h>
#include <hip/hip_bf16.h>

typedef float v2f __attribute__((ext_vector_type(2)));
typedef float v8f __attribute__((ext_vector_type(8)));

#define IN_F   128
#define OUT_F  128
#define LDSW   (2 * IN_F + 4)     // padded LDS row stride (floats) -> conflict-free
#define EPSN   1e-12f

#define WMMA_F32(a, b, c) \
    __builtin_amdgcn_wmma_f32_16x16x4_f32(false, (a), false, (b), (short)0, (c), false, false)

// ---------------------------------------------------------------- zero scratch
__global__ void zero_f4(float4* __restrict__ p, int n4) {
    int i = blockIdx.x * blockDim.x + threadIdx.x;
    int stride = gridDim.x * blockDim.x;
    float4 z = {0.f, 0.f, 0.f, 0.f};
    for (; i < n4; i += stride) p[i] = z;
}

// ------------------------------------------------- segmented SpMM (rows sorted)
// One wave per contiguous 64-edge chunk. Lane l owns features [4l, 4l+4).
// Edge metadata preloaded coalesced (2 regs/lane), broadcast via __shfl with
// compile-time lane indices; x-row gather pipelined one edge ahead. Interior
// rows are chunk-exclusive -> plain b128 store; boundary rows -> f32 atomics.
__global__ void spmm_seg(const float* __restrict__ x,
                         const int*   __restrict__ rows,
                         const int*   __restrict__ cols,
                         const float* __restrict__ vals,
                         float* __restrict__ nb, int E) {
    const int lane = threadIdx.x & 31;
    const int wave = blockIdx.x * (blockDim.x >> 5) + (threadIdx.x >> 5);
    int s = wave * 64;
    if (s >= E) return;
    const int f = lane * 4;

    int e0 = s + lane;      if (e0 > E - 1) e0 = E - 1;
    int e1 = s + 32 + lane; if (e1 > E - 1) e1 = E - 1;
    const int   rA = rows[e0], rB = rows[e1];
    const int   cA = cols[e0], cB = cols[e1];
    const float vA = (s + lane      < E) ? vals[e0] : 0.f;  // padded edges add 0
    const float vB = (s + 32 + lane < E) ? vals[e1] : 0.f;

    float ax = 0.f, ay = 0.f, az = 0.f, aw = 0.f;
    const int firstRow = __shfl(rA, 0);
    int curRow = firstRow;

    int c_next = __shfl(cA, 0);
    float4 xv_next = *(const float4*)(x + (size_t)c_next * IN_F + f);

    #pragma unroll
    for (int j = 0; j < 64; ++j) {
        const int   r = __shfl((j < 32) ? rA : rB, j & 31);   // uniform
        const float v = __shfl((j < 32) ? vA : vB, j & 31);
        float4 xv = xv_next;
        if (j < 63) {
            c_next = __shfl((j + 1 < 32) ? cA : cB, (j + 1) & 31);
            xv_next = *(const float4*)(x + (size_t)c_next * IN_F + f);
        }
        if (r != curRow) {
            float* d = nb + (size_t)curRow * IN_F + f;
            if (curRow == firstRow) {          // chunk-boundary row: atomic
                atomicAdd(d + 0, ax); atomicAdd(d + 1, ay);
                atomicAdd(d + 2, az); atomicAdd(d + 3, aw);
            } else {                           // interior row: exclusive owner
                float4 v4 = {ax, ay, az, aw};
                *(float4*)d = v4;
            }
            ax = ay = az = aw = 0.f;
            curRow = r;
        }
        ax = fmaf(v, xv.x, ax); ay = fmaf(v, xv.y, ay);
        az = fmaf(v, xv.z, az); aw = fmaf(v, xv.w, aw);
    }
    float* d = nb + (size_t)curRow * IN_F + f; // row may continue in next chunk
    atomicAdd(d + 0, ax); atomicAdd(d + 1, ay);
    atomicAdd(d + 2, az); atomicAdd(d + 3, aw);
}

// --------------------------- fused [x|nb] @ W^T + b, row-L2-normalize, via WMMA
// Each wave computes a 32x128 strip (two M-tiles share every B fragment).
// K-loop split into two fully-unrolled halves with fixed base pointers so all
// global/LDS loads use immediate offsets. Per k-step: 2 global b64 A-loads
// (prefetched 1 step ahead), 8 immediate-offset ds_load_b64 into live regs,
// then 16 independent v_wmma_f32_16x16x4_f32.
__global__ void __launch_bounds__(256)
gemm_norm(const float* __restrict__ x, const float* __restrict__ nb,
          const float* __restrict__ W, const float* __restrict__ bias,
          float* __restrict__ out, int nRows, int nTiles) {
    extern __shared__ float sW[];   // [OUT_F][LDSW] padded, 130 KB

    // Cooperative staged load of W (row-major [OUT_F][2*IN_F]) into padded LDS.
    {
        const float4* Wv = (const float4*)W;
        for (int i = threadIdx.x; i < OUT_F * (2 * IN_F / 4); i += blockDim.x) {
            int n = i >> 6;             // 64 float4 per W row
            int k4 = i & 63;
            float4 w4 = Wv[i];
            float* dp = sW + n * LDSW + (k4 << 2);
            dp[0] = w4.x; dp[1] = w4.y; dp[2] = w4.z; dp[3] = w4.w;
        }
    }
    __syncthreads();

    const int lane = threadIdx.x & 31;
    const int wid  = threadIdx.x >> 5;
    const int nJobs = (nTiles + 1) >> 1;
    const int job = blockIdx.x * 8 + wid;      // one job = two consecutive M-tiles
    if (job >= nJobs) return;
    const int tile0 = job * 2;
    const int tile1 = tile0 + 1;

    const int ln = lane & 15;
    const int g  = lane >> 4;                   // half-wave group
    const int koff = g * 2;
    int r0i = tile0 * 16 + ln; if (r0i >= nRows) r0i = nRows - 1;
    int r1i = tile1 * 16 + ln; if (r1i >= nRows) r1i = nRows - 1;
    const float* pa0 = x  + (size_t)r0i * IN_F + koff;   // A bases, half 0
    const float* pa1 = x  + (size_t)r1i * IN_F + koff;
    const float* pn0 = nb + (size_t)r0i * IN_F + koff;   // A bases, half 1
    const float* pn1 = nb + (size_t)r1i * IN_F + koff;
    // two LDS bases so every fragment offset fits the 16-bit DS immediate
    const float* sbA = sW + ln * LDSW + koff;            // nt = 0..3
    const float* sbB = sbA + 4 * (16 * LDSW);            // nt = 4..7

    v8f acc0[8], acc1[8];
    #pragma unroll
    for (int nt = 0; nt < 8; ++nt)
        #pragma unroll
        for (int r = 0; r < 8; ++r) { acc0[nt][r] = 0.f; acc1[nt][r] = 0.f; }

    v2f a0n = *(const v2f*)(pa0);
    v2f a1n = *(const v2f*)(pa1);

    // ---- half 0: K = 0..127 (A from x) ----
    #pragma unroll
    for (int kk = 0; kk < 32; ++kk) {
        v2f a0 = a0n, a1 = a1n;
        if (kk < 31) {                                  // prefetch next A frags
            a0n = *(const v2f*)(pa0 + (kk + 1) * 4);
            a1n = *(const v2f*)(pa1 + (kk + 1) * 4);
        } else {                                        // cross into nb half
            a0n = *(const v2f*)(pn0);
            a1n = *(const v2f*)(pn1);
        }
        v2f bf[8];
        #pragma unroll
        for (int nt = 0; nt < 8; ++nt)                  // immediate-offset ds_load
            bf[nt] = *(const v2f*)(((nt < 4) ? sbA : sbB) + (nt & 3) * (16 * LDSW) + kk * 4);
        #pragma unroll
        for (int nt = 0; nt < 8; ++nt) {                // 16 independent WMMAs
            acc0[nt] = WMMA_F32(a0, bf[nt], acc0[nt]);
            acc1[nt] = WMMA_F32(a1, bf[nt], acc1[nt]);
        }
    }
    // ---- half 1: K = 128..255 (A from nb) ----
    #pragma unroll
    for (int kk = 0; kk < 32; ++kk) {
        v2f a0 = a0n, a1 = a1n;
        if (kk < 31) {
            a0n = *(const v2f*)(pn0 + (kk + 1) * 4);
            a1n = *(const v2f*)(pn1 + (kk + 1) * 4);
        }
        v2f bf[8];
        #pragma unroll
        for (int nt = 0; nt < 8; ++nt)
            bf[nt] = *(const v2f*)(((nt < 4) ? sbA : sbB) + (nt & 3) * (16 * LDSW) + IN_F + kk * 4);
        #pragma unroll
        for (int nt = 0; nt < 8; ++nt) {
            acc0[nt] = WMMA_F32(a0, bf[nt], acc0[nt]);
            acc1[nt] = WMMA_F32(a1, bf[nt], acc1[nt]);
        }
    }

    // bias add
    #pragma unroll
    for (int nt = 0; nt < 8; ++nt) {
        const float bv = bias[nt * 16 + ln];
        #pragma unroll
        for (int r = 0; r < 8; ++r) { acc0[nt][r] += bv; acc1[nt][r] += bv; }
    }

    // per-row 1/max(||.||,eps): reduce squares over the 16 lanes of this half
    float sc0[8], sc1[8];
    #pragma unroll
    for (int r = 0; r < 8; ++r) {
        float s0 = 0.f, s1 = 0.f;
        #pragma unroll
        for (int nt = 0; nt < 8; ++nt) {
            s0 = fmaf(acc0[nt][r], acc0[nt][r], s0);
            s1 = fmaf(acc1[nt][r], acc1[nt][r], s1);
        }
        s0 += __shfl_xor(s0, 1); s1 += __shfl_xor(s1, 1);
        s0 += __shfl_xor(s0, 2); s1 += __shfl_xor(s1, 2);
        s0 += __shfl_xor(s0, 4); s1 += __shfl_xor(s1, 4);
        s0 += __shfl_xor(s0, 8); s1 += __shfl_xor(s1, 8);  // within 16-lane group
        sc0[r] = 1.0f / fmaxf(sqrtf(s0), EPSN);
        sc1[r] = 1.0f / fmaxf(sqrtf(s1), EPSN);
    }

    // scaled stores (C/D layout: VGPR r, lane -> row r + 8*g, col nt*16 + ln)
    #pragma unroll
    for (int nt = 0; nt < 8; ++nt)
        #pragma unroll
        for (int r = 0; r < 8; ++r) {
            const int M = (g << 3) + r;
            int o0 = tile0 * 16 + M;
            int o1 = tile1 * 16 + M;
            if (o0 < nRows) out[(size_t)o0 * OUT_F + nt * 16 + ln] = acc0[nt][r] * sc0[r];
            if (o1 < nRows) out[(size_t)o1 * OUT_F + nt * 16 + ln] = acc1[nt][r] * sc1[r];
        }
}

// ----------------------------------------------------------------------- launch
extern "C" void kernel_launch(void* const* d_in, const int* in_sizes, int n_in,
                              void* d_out, int out_size, void* d_ws, size_t ws_size,
                              hipStream_t stream) {
    const float* x    = (const float*)d_in[0];
    const int*   rows = (const int*)  d_in[1];
    const int*   cols = (const int*)  d_in[2];
    const float* vals = (const float*)d_in[3];
    const float* W    = (const float*)d_in[4];
    const float* bias = (const float*)d_in[5];
    float* out  = (float*)d_out;
    float* nbuf = (float*)d_ws;     // neighbor accumulator [N][IN_F]

    const int E  = in_sizes[1];
    const int Nn = in_sizes[0] / IN_F;

    // 1) zero neighbor buffer
    int n4 = Nn * IN_F / 4;
    int zb = (n4 + 255) / 256; if (zb > 2048) zb = 2048;
    zero_f4<<<zb, 256, 0, stream>>>((float4*)nbuf, n4);

    // 2) segmented SpMM: one wave per 64-edge chunk, 8 waves / block
    int waves = (E + 63) / 64;
    int blocks1 = (waves + 7) / 8;
    spmm_seg<<<blocks1, 256, 0, stream>>>(x, rows, cols, vals, nbuf, E);

    // 3) fused WMMA GEMM + bias + L2 normalize; 8 jobs (16 M-tiles) per block
    int nTiles = (Nn + 15) / 16;
    int nJobs  = (nTiles + 1) / 2;
    int blocks2 = (nJobs + 7) / 8;
    size_t shmem = (size_t)OUT_F * LDSW * sizeof(float);   // 133,120 B
    hipFuncSetAttribute(reinterpret_cast<const void*>(gemm_norm),
                        hipFuncAttributeMaxDynamicSharedMemorySize, (int)shmem);
    gemm_norm<<<blocks2, 256, shmem, stream>>>(x, nbuf, W, bias, out, Nn, nTiles);
}